// OutputModule_53154515256155
// MI455X (gfx1250) — compile-verified
//
#include <hip/hip_runtime.h>

#define BATCH      256
#define SEQ        128
#define HIDDEN     1024
#define NUM_BLOCKS 256
#define NUM_LABELS 3

typedef __attribute__((ext_vector_type(2))) float v2f;
typedef __attribute__((ext_vector_type(8))) float v8f;

// ---------------------------------------------------------------------------
// Kernel 1: one workgroup per k (256 blocks, 256 threads = 8 waves).
//   scores[b] = dot(x[b,:], S[b,k,:])   (coalesced float4 loads, wave reduce)
//   p = softmax_b(scores)               (LDS tree reduce)
//   u[k,h] = sum_b p[b]*S[b,k,h]        (second pass; S slice is L2-resident)
// ---------------------------------------------------------------------------
__global__ __launch_bounds__(256) void k1_scores_softmax_u(
    const float* __restrict__ ent,     // [B][SEQ][HIDDEN], x = ent[:,0,:]
    const float* __restrict__ states,  // [B][NUM_BLOCKS][HIDDEN]
    float* __restrict__ u)             // [NUM_BLOCKS][HIDDEN]
{
    const int k    = blockIdx.x;
    const int t    = threadIdx.x;
    const int wave = t >> 5;
    const int lane = t & 31;

    __shared__ float sc[BATCH];    // scores, then p
    __shared__ float red[256];

    // ---- Phase A: scores ----
    for (int i = 0; i < 32; ++i) {
        const int b = wave * 32 + i;
        const float* xs = ent    + (size_t)b * SEQ * HIDDEN;
        const float* ss = states + ((size_t)b * NUM_BLOCKS + k) * HIDDEN;
        float partial = 0.f;
        #pragma unroll
        for (int j = 0; j < HIDDEN / 128; ++j) {          // 8 iters of float4
            const int h = (j * 32 + lane) * 4;
            const float4 xv = *(const float4*)(xs + h);
            const float4 sv = *(const float4*)(ss + h);
            partial += xv.x * sv.x + xv.y * sv.y + xv.z * sv.z + xv.w * sv.w;
        }
        #pragma unroll
        for (int off = 16; off >= 1; off >>= 1)
            partial += __shfl_xor(partial, off, 32);
        if (lane == 0) sc[b] = partial;
    }
    __syncthreads();

    // ---- Phase B: softmax over b ----
    const float myscore = sc[t];
    red[t] = myscore;
    __syncthreads();
    for (int s = 128; s >= 1; s >>= 1) {
        if (t < s) red[t] = fmaxf(red[t], red[t + s]);
        __syncthreads();
    }
    const float mx = red[0];
    __syncthreads();
    const float e = __expf(myscore - mx);
    red[t] = e;
    __syncthreads();
    for (int s = 128; s >= 1; s >>= 1) {
        if (t < s) red[t] += red[t + s];
        __syncthreads();
    }
    const float denom = red[0];
    __syncthreads();
    sc[t] = e / denom;
    __syncthreads();

    // ---- Phase C: u[k,:] = sum_b p[b] * S[b,k,:] ----
    const int h0 = t * 4;
    float4 acc = make_float4(0.f, 0.f, 0.f, 0.f);
    for (int b = 0; b < BATCH; ++b) {
        const float pb = sc[b];
        const float4 sv = *(const float4*)(states +
            ((size_t)b * NUM_BLOCKS + k) * HIDDEN + h0);
        acc.x += pb * sv.x; acc.y += pb * sv.y;
        acc.z += pb * sv.z; acc.w += pb * sv.w;
    }
    *(float4*)(u + (size_t)k * HIDDEN + h0) = acc;
}

// ---------------------------------------------------------------------------
// Kernel 2: v = x + u @ H_w^T  via V_WMMA_F32_16X16X4_F32, fused PReLU.
// One wave computes one 16x16 tile of the (256 x 1024) result.
// 1024 waves = 128 blocks x 8 waves.
// A-layout (16x4 f32): lane = {m = lane&15, k-half = lane>>4}, 2 VGPRs.
// C-layout: VGPR j holds row (j + 8*(lane>>4)), col = lane&15.
// ---------------------------------------------------------------------------
__global__ __launch_bounds__(256) void k2_gemm_prelu(
    const float* __restrict__ u,      // [256][HIDDEN]
    const float* __restrict__ Hw,     // [HIDDEN][HIDDEN]
    const float* __restrict__ ent,    // x = ent[:,0,:]
    const float* __restrict__ alpha,  // [HIDDEN]
    float* __restrict__ act)          // [256][HIDDEN]
{
    const int t    = threadIdx.x;
    const int lane = t & 31;
    const int w    = blockIdx.x * 8 + (t >> 5);   // global wave id, 0..1023
    const int mt   = w & 15;                      // 16 tiles over rows (256)
    const int nt   = w >> 4;                      // 64 tiles over cols (1024)

    const int m    = lane & 15;
    const int half = lane >> 4;                   // 0 or 1
    const int kb   = half * 2;                    // K offset for this lane half

    const float* arow = u  + (size_t)(mt * 16 + m) * HIDDEN + kb;  // A: u rows
    const float* brow = Hw + (size_t)(nt * 16 + m) * HIDDEN + kb;  // B: Hw rows (n = lane&15)

    v8f c = {};
    for (int k0 = 0; k0 < HIDDEN; k0 += 4) {
        const v2f a = *(const v2f*)(arow + k0);
        const v2f b = *(const v2f*)(brow + k0);
        c = __builtin_amdgcn_wmma_f32_16x16x4_f32(
                /*neg_a=*/false, a, /*neg_b=*/false, b,
                /*c_mod=*/(short)0, c, /*reuse_a=*/false, /*reuse_b=*/false);
    }

    // Epilogue: v = c + x, PReLU, store act.
    const int col = nt * 16 + (lane & 15);
    const float al = alpha[col];
    const int rbase = mt * 16 + half * 8;
    #pragma unroll
    for (int j = 0; j < 8; ++j) {
        const int row = rbase + j;
        const float xv = ent[(size_t)row * SEQ * HIDDEN + col];
        const float vv = c[j] + xv;
        act[(size_t)row * HIDDEN + col] = (vv >= 0.f) ? vv : al * vv;
    }
}

// ---------------------------------------------------------------------------
// Kernel 3: out = act @ R_w^T  (256 x 3). One block per output row.
// ---------------------------------------------------------------------------
__global__ __launch_bounds__(256) void k3_labels(
    const float* __restrict__ act,   // [256][HIDDEN]
    const float* __restrict__ Rw,    // [NUM_LABELS][HIDDEN]
    float* __restrict__ out)         // [256][NUM_LABELS]
{
    const int b = blockIdx.x;
    const int t = threadIdx.x;
    __shared__ float red[NUM_LABELS][256];

    const float4 av = *(const float4*)(act + (size_t)b * HIDDEN + t * 4);
    const float4 r0 = *(const float4*)(Rw + 0 * HIDDEN + t * 4);
    const float4 r1 = *(const float4*)(Rw + 1 * HIDDEN + t * 4);
    const float4 r2 = *(const float4*)(Rw + 2 * HIDDEN + t * 4);

    red[0][t] = av.x*r0.x + av.y*r0.y + av.z*r0.z + av.w*r0.w;
    red[1][t] = av.x*r1.x + av.y*r1.y + av.z*r1.z + av.w*r1.w;
    red[2][t] = av.x*r2.x + av.y*r2.y + av.z*r2.z + av.w*r2.w;
    __syncthreads();

    for (int s = 128; s >= 1; s >>= 1) {
        if (t < s) {
            red[0][t] += red[0][t + s];
            red[1][t] += red[1][t + s];
            red[2][t] += red[2][t + s];
        }
        __syncthreads();
    }
    if (t < NUM_LABELS) out[b * NUM_LABELS + t] = red[t][0];
}

// ---------------------------------------------------------------------------
extern "C" void kernel_launch(void* const* d_in, const int* in_sizes, int n_in,
                              void* d_out, int out_size, void* d_ws, size_t ws_size,
                              hipStream_t stream) {
    const float* ent    = (const float*)d_in[0];  // entity_encoding [B][SEQ][HIDDEN]
    const float* states = (const float*)d_in[1];  // states [B][NUM_BLOCKS*HIDDEN]
    const float* Hw     = (const float*)d_in[2];  // H_w [HIDDEN][HIDDEN]
    const float* Rw     = (const float*)d_in[3];  // R_w [NUM_LABELS][HIDDEN]
    const float* alpha  = (const float*)d_in[4];  // prelu_alpha [HIDDEN]
    float* out = (float*)d_out;

    float* u   = (float*)d_ws;                                   // 1 MB
    float* act = (float*)((char*)d_ws + (size_t)NUM_BLOCKS * HIDDEN * sizeof(float));

    k1_scores_softmax_u<<<NUM_BLOCKS, 256, 0, stream>>>(ent, states, u);
    k2_gemm_prelu<<<128, 256, 0, stream>>>(u, Hw, ent, alpha, act);
    k3_labels<<<BATCH, 256, 0, stream>>>(act, Rw, out);
}